// DeltaNet_31877247271497
// MI455X (gfx1250) — compile-verified
//
#include <hip/hip_runtime.h>
#include <math.h>
#include <stdint.h>

typedef __attribute__((ext_vector_type(2))) float v2f;
typedef __attribute__((ext_vector_type(8))) float v8f;

#define LDP 132   // padded stride for 32x128 / 128x128 tiles (132 % 64 == 4 -> conflict free)
#define LTP 34    // padded stride for 32x32 T tile (even -> v2f aligned)

// ---- CDNA5 async global -> LDS copy (ASYNCcnt tracked) ----
__device__ __forceinline__ unsigned lds_off32(const void* p) {
  // generic LDS pointer: low 32 bits = byte offset within workgroup LDS allocation
  return (unsigned)(uintptr_t)p;
}
__device__ __forceinline__ void async_ld_b128(unsigned lds, const float* g) {
  asm volatile("global_load_async_to_lds_b128 %0, %1, off"
               : : "v"(lds), "v"(g) : "memory");
}
__device__ __forceinline__ void async_ld_b32(unsigned lds, const float* g) {
  asm volatile("global_load_async_to_lds_b32 %0, %1, off"
               : : "v"(lds), "v"(g) : "memory");
}
__device__ __forceinline__ void wait_async0() {
#if __has_builtin(__builtin_amdgcn_s_wait_asynccnt)
  __builtin_amdgcn_s_wait_asynccnt(0);
#else
  asm volatile("s_wait_asynccnt 0x0" ::: "memory");
#endif
}

// C(16x16) += A(16xK) * B(Kx16), operands in LDS (fp32 WMMA 16x16x4).
// A element (m,k): AT ? pA[k*lda + m] : pA[m*lda + k]
// B element (k,n): BT ? pB[n*ldb + k] : pB[k*ldb + n]
template<bool AT, bool BT>
__device__ __forceinline__ v8f mm16(const float* pA, int lda,
                                    const float* pB, int ldb,
                                    int K, v8f acc) {
  const int lane = threadIdx.x & 31;
  const int lo   = lane & 15;
  const int kk   = (lane >> 4) << 1;   // 0 or 2 (K split across lane halves)
#pragma unroll
  for (int k0 = 0; k0 < K; k0 += 4) {
    const int ka = k0 + kk;
    v2f a, b;
    if (AT) { a.x = pA[(ka + 0) * lda + lo]; a.y = pA[(ka + 1) * lda + lo]; }
    else    { a = *(const v2f*)(pA + lo * lda + ka); }
    if (BT) { b = *(const v2f*)(pB + lo * ldb + ka); }
    else    { b.x = pB[(ka + 0) * ldb + lo]; b.y = pB[(ka + 1) * ldb + lo]; }
    acc = __builtin_amdgcn_wmma_f32_16x16x4_f32(false, a, false, b,
                                                (short)0, acc, false, false);
  }
  return acc;
}

__device__ __forceinline__ void store_tile(float* pC, int ldc, v8f c) {
  const int lane = threadIdx.x & 31;
  const int lo = lane & 15, hi = lane >> 4;
#pragma unroll
  for (int r = 0; r < 8; ++r) pC[(r + 8 * hi) * ldc + lo] = c[r];
}

__global__ __launch_bounds__(512)
void deltanet_fused_kernel(const float* __restrict__ q,
                           const float* __restrict__ k,
                           const float* __restrict__ v,
                           const float* __restrict__ beta,
                           float* __restrict__ o,
                           float* __restrict__ Sout,
                           int L) {
  // double-buffered chunk inputs (async-prefetched), single-buffer work tiles
  __shared__ float sQ[2][32 * LDP];
  __shared__ float sK[2][32 * LDP];
  __shared__ float sV[2][32 * LDP];
  __shared__ float sU[32 * LDP];
  __shared__ float sW[32 * LDP];
  __shared__ float sS[128 * LDP];
  __shared__ float sT[32 * LTP];
  __shared__ float sBeta[2][32];

  const int bh   = blockIdx.x;
  const int tid  = threadIdx.x;
  const int wave = tid >> 5;
  const int lane = tid & 31;
  const int lo   = lane & 15, hi = lane >> 4;
  const int nchunks = L / 32;
  const size_t base = (size_t)bh * L * 128;

  // issue async prefetch of chunk `chn`'s q,k,v,beta into buffer parity `pn`
  auto prefetch = [&](int chn, int pn) {
    const int t0n = chn * 32;
    for (int idx = tid; idx < 3 * 1024; idx += 512) {     // 3 matrices x 1024 float4
      const int m   = idx >> 10;
      const int r4  = idx & 1023;
      const int row = r4 >> 5;
      const int c4  = (r4 & 31) << 2;
      const float* src = (m == 0) ? q : (m == 1) ? k : v;
      float* dst = ((m == 0) ? sQ[pn] : (m == 1) ? sK[pn] : sV[pn]) + row * LDP + c4;
      async_ld_b128(lds_off32(dst), src + base + (size_t)(t0n + row) * 128 + c4);
    }
    if (tid < 32)
      async_ld_b32(lds_off32(&sBeta[pn][tid]), beta + (size_t)bh * L + t0n + tid);
  };

  // prologue: fetch chunk 0 while zeroing S
  prefetch(0, 0);
  for (int idx = tid; idx < 128 * LDP; idx += 512) sS[idx] = 0.f;
  wait_async0();
  __syncthreads();

  for (int ch = 0; ch < nchunks; ++ch) {
    const int t0 = ch * 32;
    const int p  = ch & 1;
    float* cQ = sQ[p];
    float* cK = sK[p];
    float* cV = sV[p];
    float* cB = sBeta[p];

    // ---- kick off async prefetch of the next chunk into the other buffer ----
    if (ch + 1 < nchunks) prefetch(ch + 1, 1 - p);

    // ---- Phase 2: l2-normalize rows of q,k (wave handles 2 rows) ----
#pragma unroll
    for (int rr = 0; rr < 2; ++rr) {
      const int row = wave * 2 + rr;
      float sq = 0.f, sk = 0.f;
      for (int dd = lane; dd < 128; dd += 32) {
        const float a = cQ[row * LDP + dd]; sq += a * a;
        const float b = cK[row * LDP + dd]; sk += b * b;
      }
#pragma unroll
      for (int off = 16; off; off >>= 1) {
        sq += __shfl_xor(sq, off, 32);
        sk += __shfl_xor(sk, off, 32);
      }
      const float iq = 1.f / sqrtf(sq + 1e-6f);
      const float ik = 1.f / sqrtf(sk + 1e-6f);
      for (int dd = lane; dd < 128; dd += 32) {
        cQ[row * LDP + dd] *= iq;
        cK[row * LDP + dd] *= ik;
      }
    }
    __syncthreads();

    // ---- Phase 3: M = -(beta_i) * (K K^T), strictly lower -> sT ----
    if (wave < 4) {
      const int ti = wave >> 1, tj = wave & 1;
      v8f c = {};
      c = mm16<false, true>(cK + ti * 16 * LDP, LDP, cK + tj * 16 * LDP, LDP, 128, c);
#pragma unroll
      for (int r = 0; r < 8; ++r) {
        const int i = ti * 16 + r + 8 * hi, j = tj * 16 + lo;
        sT[i * LTP + j] = (j < i) ? (-cB[i] * c[r]) : 0.f;
      }
    }
    __syncthreads();

    // ---- Phase 4: forward substitution T=(I-M)^-1, then fold beta into columns ----
    if (wave == 0) {
      for (int i = 1; i < 32; ++i) {
        float sum = 0.f;
        for (int j = 0; j < i; ++j) sum += sT[i * LTP + j] * sT[j * LTP + lane];
        sT[i * LTP + lane] += sum;
      }
#pragma unroll
      for (int i = 0; i < 32; ++i) {
        const float t = sT[i * LTP + lane] + ((i == lane) ? 1.f : 0.f);
        sT[i * LTP + lane] = t * cB[lane];   // column scale: absorbs beta for u,w
      }
    }
    __syncthreads();

    // ---- Phase 5: u = T'@V (raw v), w = T'@K  (32 tile jobs) ----
    for (int job = wave; job < 32; job += 16) {
      const int which = job >> 4, t = job & 15, ti = t >> 3, tj = t & 7;
      v8f c = {};
      c = mm16<false, false>(sT + ti * 16 * LTP, LTP,
                             (which ? cK : cV) + tj * 16, LDP, 32, c);
      store_tile((which ? sW : sU) + ti * 16 * LDP + tj * 16, LDP, c);
    }
    __syncthreads();

    // ---- Phase 6: A_loc = Q@K^T (masked, -> sT) ; u -= w@S  (20 tile jobs) ----
    for (int job = wave; job < 20; job += 16) {
      if (job < 4) {
        const int ti = job >> 1, tj = job & 1;
        v8f c = {};
        c = mm16<false, true>(cQ + ti * 16 * LDP, LDP, cK + tj * 16 * LDP, LDP, 128, c);
#pragma unroll
        for (int r = 0; r < 8; ++r) {
          const int i = ti * 16 + r + 8 * hi, j = tj * 16 + lo;
          sT[i * LTP + j] = (j <= i) ? c[r] : 0.f;
        }
      } else {
        const int t = job - 4, ti = t >> 3, tj = t & 7;
        v8f c = {};
        c = mm16<false, false>(sW + ti * 16 * LDP, LDP, sS + tj * 16, LDP, 128, c);
        float* pu = sU + ti * 16 * LDP + tj * 16;
#pragma unroll
        for (int r = 0; r < 8; ++r) pu[(r + 8 * hi) * LDP + lo] -= c[r];
      }
    }
    __syncthreads();

    // ---- Phase 7: o = Q@S + A_loc@u -> global (16 tile jobs) ----
    for (int job = wave; job < 16; job += 16) {
      const int ti = job >> 3, tj = job & 7;
      v8f c = {};
      c = mm16<false, false>(cQ + ti * 16 * LDP, LDP, sS + tj * 16, LDP, 128, c);
      c = mm16<false, false>(sT + ti * 16 * LTP, LTP, sU + tj * 16, LDP, 32, c);
      float* po = o + base + (size_t)(t0 + ti * 16) * 128 + tj * 16;
#pragma unroll
      for (int r = 0; r < 8; ++r) po[(size_t)(r + 8 * hi) * 128 + lo] = c[r];
    }
    __syncthreads();

    // ---- Phase 8: S += K^T @ u  (64 tile jobs) ----
    for (int job = wave; job < 64; job += 16) {
      const int ti = job >> 3, tj = job & 7;
      v8f c = {};
      c = mm16<true, false>(cK + ti * 16, LDP, sU + tj * 16, LDP, 32, c);
      float* ps = sS + ti * 16 * LDP + tj * 16;
#pragma unroll
      for (int r = 0; r < 8; ++r) ps[(r + 8 * hi) * LDP + lo] += c[r];
    }

    // ---- step boundary: next chunk's async data must be fully in LDS ----
    wait_async0();
    __syncthreads();
  }

  // ---- Final: write S (b,h,dk,dv) ----
  float* So = Sout + (size_t)bh * 128 * 128;
  for (int idx = tid; idx < 128 * 128; idx += 512)
    So[idx] = sS[(idx >> 7) * LDP + (idx & 127)];
}

extern "C" void kernel_launch(void* const* d_in, const int* in_sizes, int n_in,
                              void* d_out, int out_size, void* d_ws, size_t ws_size,
                              hipStream_t stream) {
  (void)n_in; (void)out_size; (void)d_ws; (void)ws_size;
  const float* q    = (const float*)d_in[0];
  const float* k    = (const float*)d_in[1];
  const float* v    = (const float*)d_in[2];
  const float* beta = (const float*)d_in[3];
  // q: (b,h,L,128), beta: (b,h,L)
  const int L  = 4096;
  const int BH = in_sizes[3] / L;   // b*h = 16
  float* o    = (float*)d_out;
  float* Sout = o + (size_t)BH * L * 128;
  deltanet_fused_kernel<<<BH, 512, 0, stream>>>(q, k, v, beta, o, Sout, L);
}